// BOJANET_72284299591834
// MI455X (gfx1250) — compile-verified
//
#include <hip/hip_runtime.h>
#include <math.h>

#define B_TOT 256
#define SEQ   4096
#define HID   16
#define VD    6
#define WIN   16
#define NBLK  16          // 256 batches / 16 per WMMA tile

typedef __attribute__((ext_vector_type(2))) float v2f;
typedef __attribute__((ext_vector_type(8))) float v8f;

#define LOG2E  1.44269504088896340736f

#if __has_builtin(__builtin_amdgcn_tanhf)
#define TANHF(xx) __builtin_amdgcn_tanhf(xx)
#else
#define TANHF(xx) (1.f - 2.f * __builtin_amdgcn_rcpf(1.f + __builtin_amdgcn_exp2f(2.f * LOG2E * (xx))))
#endif

// ---------------------------------------------------------------------------
// Shared FIR helper: complex 16-tap FIR at (b, s). s is block-uniform so the
// causal-boundary branch is a scalar branch.
// ---------------------------------------------------------------------------
__device__ __forceinline__ void fir_cs(const float* __restrict__ x, int b, int s,
                                       const float* fI, const float* fQ,
                                       float* If, float* Qf) {
  float wI[WIN], wQ[WIN];
#pragma unroll
  for (int t = 0; t < WIN; ++t) {
    int si = s + t - (WIN - 1);
    float xi = 0.f, xq = 0.f;
    if (si >= 0) {
      const float* p = x + ((size_t)b * SEQ + si) * 2;
      xi = p[0];
      xq = p[1];
    }
    wI[t] = xi;
    wQ[t] = xq;
  }
#pragma unroll
  for (int v = 0; v < VD; ++v) {
    float a = 0.f, bb = 0.f, c = 0.f, d = 0.f;
#pragma unroll
    for (int t = 0; t < WIN; ++t) {
      float iw = fI[v * WIN + t];
      float qw = fQ[v * WIN + t];
      a += wI[t] * iw;   // wI * fir_I
      bb += wQ[t] * qw;  // wQ * fir_Q
      c += wI[t] * qw;   // wI * fir_Q
      d += wQ[t] * iw;   // wQ * fir_I
    }
    If[v] = a - bb;
    Qf[v] = c + d;
  }
}

// ---------------------------------------------------------------------------
// Kernel A (parallel): FIR -> magnitudes -> fi/gi gate pre-activations.
// Grid: SEQ blocks (one per time step) x 256 threads (one per batch).
// fiW/giW layout: [s][b][HID], so the scan reads contiguous 1KB tiles.
// ---------------------------------------------------------------------------
__global__ void __launch_bounds__(256) fir_gate_kernel(
    const float* __restrict__ x,
    const float* __restrict__ firI, const float* __restrict__ firQ,
    const float* __restrict__ Wfi, const float* __restrict__ bfi,
    const float* __restrict__ Wgi, const float* __restrict__ bgi,
    float* __restrict__ fiW, float* __restrict__ giW) {
  __shared__ float sFI[VD * WIN], sFQ[VD * WIN];
  __shared__ float sWf[HID * 2 * VD], sWg[HID * 2 * VD];
  __shared__ float sbf[HID], sbg[HID];
  int tid = threadIdx.x;
  if (tid < VD * WIN) { sFI[tid] = firI[tid]; sFQ[tid] = firQ[tid]; }
  if (tid < HID * 2 * VD) { sWf[tid] = Wfi[tid]; sWg[tid] = Wgi[tid]; }
  if (tid < HID) { sbf[tid] = bfi[tid]; sbg[tid] = bgi[tid]; }
  __syncthreads();

  int s = blockIdx.x;
  int b = tid;

  float If[VD], Qf[VD];
  fir_cs(x, b, s, sFI, sFQ, If, Qf);

  float mg[VD], mg2[VD];
#pragma unroll
  for (int v = 0; v < VD; ++v) {
    float m = sqrtf(If[v] * If[v] + Qf[v] * Qf[v]) + 1e-8f;
    mg[v] = m;
    mg2[v] = m * m;
  }

  float* fo = fiW + ((size_t)s * B_TOT + b) * HID;
  float* go = giW + ((size_t)s * B_TOT + b) * HID;
#pragma unroll
  for (int h = 0; h < HID; ++h) {
    float af = sbf[h], ag = sbg[h];
#pragma unroll
    for (int v = 0; v < VD; ++v) {
      af += mg[v] * sWf[h * (2 * VD) + v] + mg2[v] * sWf[h * (2 * VD) + VD + v];
      ag += mg[v] * sWg[h * (2 * VD) + v] + mg2[v] * sWg[h * (2 * VD) + VD + v];
    }
    fo[h] = af;
    go[h] = ag;
  }
}

// ---------------------------------------------------------------------------
// Kernel B (sequential scan): one wave per block of 16 batch elements.
// Per step: D[j,b] = sum_k W[j,k] * h[b,k] via 4 chained
// V_WMMA_F32_16X16X4_F32 per gate (A = weights, constant; B = h^T).
//
// Shuffle-free trick: permute the contracted (K) axis of W when loading A so
// that the C/D-layout registers of h are *directly* the B operands:
//   perm = {0,1,8,9, 2,3,10,11, 4,5,12,13, 6,7,14,15}
//   A[p][q] = W[p][perm[q]]  =>  B_k = (hD[2k], hD[2k+1]) on BOTH lane halves.
// D positions remain the logical hidden indices, so fi/gi/h0/hW are untouched.
//
// Gate tiles are software-pipelined one step ahead so their load latency is
// covered by the previous step's WMMA/tanh work instead of sitting on the
// 4096-step serial chain.
// ---------------------------------------------------------------------------
__global__ void __launch_bounds__(32) scan_kernel(
    const float* __restrict__ h0,
    const float* __restrict__ Wfh, const float* __restrict__ Wgh,
    const float* __restrict__ fiW, const float* __restrict__ giW,
    float* __restrict__ hW) {
  const int blk = blockIdx.x;       // 0..15
  const int l = threadIdx.x;        // 0..31
  const int n = l & 15;             // A row (M=j) / C-D column (N=batch)
  const bool hi = l >= 16;
  const int koff = hi ? 2 : 0;

  // K-axis permutation aligning D layout with B layout
  const int perm[16] = {0, 1, 8, 9, 2, 3, 10, 11, 4, 5, 12, 13, 6, 7, 14, 15};

  // A operands (M=j rows, K=permuted cols), constant across all 4096 steps.
  v2f Af[4], Ag[4];
#pragma unroll
  for (int k = 0; k < 4; ++k) {
    int q0 = perm[4 * k + koff];
    int q1 = perm[4 * k + koff + 1];
    Af[k].x = Wfh[n * HID + q0];
    Af[k].y = Wfh[n * HID + q1];
    Ag[k].x = Wgh[n * HID + q0];
    Ag[k].y = Wgh[n * HID + q1];
  }

  // h in C/D layout: lane l holds h[j = r + 8*(l>=16)][b = l%16], r=0..7
  float hD[8];
  {
    const float* hp0 = h0 + ((size_t)(blk * 16 + n)) * HID + (hi ? 8 : 0);
#pragma unroll
    for (int r = 0; r < 8; ++r) hD[r] = hp0[r];
  }

  const size_t baseElem = ((size_t)blk * 16 + n) * HID + (hi ? 8 : 0);
  const size_t stride = (size_t)B_TOT * HID;  // floats per step
  const float* fip = fiW + baseElem;
  const float* gip = giW + baseElem;
  float* hp = hW + baseElem;

  // ---- software-pipeline prologue: tiles for step 0 ----
  v8f cfN, cgN;
#pragma unroll
  for (int r = 0; r < 8; ++r) {
    cfN[r] = fip[r];
    cgN[r] = gip[r];
  }

  for (int s = 0; s < SEQ; ++s) {
    const size_t ofs = (size_t)s * stride;

    // current step's tiles (loaded one iteration ago)
    v8f cf = cfN, cg = cgN;

    // issue next step's tile loads now (clamped index -> branch-free);
    // the wait for them lands in the NEXT iteration.
    const size_t ofsN = (size_t)((s + 1 < SEQ) ? s + 1 : s) * stride;
    __builtin_prefetch(fip + ofs + 2 * stride, 0, 1);
    __builtin_prefetch(gip + ofs + 2 * stride, 0, 1);
#pragma unroll
    for (int r = 0; r < 8; ++r) {
      cfN[r] = fip[ofsN + r];
      cgN[r] = gip[ofsN + r];
    }

    // B operands come straight out of the h state registers (no shuffles)
    v2f B0, B1, B2, B3;
    B0.x = hD[0]; B0.y = hD[1];
    B1.x = hD[2]; B1.y = hD[3];
    B2.x = hD[4]; B2.y = hD[5];
    B3.x = hD[6]; B3.y = hD[7];

    // cf += W_fh x h^T ; cg += W_gh x h^T  (K=16 as 4 k-slices, full f32)
    cf = __builtin_amdgcn_wmma_f32_16x16x4_f32(false, Af[0], false, B0, (short)0, cf, false, false);
    cg = __builtin_amdgcn_wmma_f32_16x16x4_f32(false, Ag[0], false, B0, (short)0, cg, false, false);
    cf = __builtin_amdgcn_wmma_f32_16x16x4_f32(false, Af[1], false, B1, (short)0, cf, false, false);
    cg = __builtin_amdgcn_wmma_f32_16x16x4_f32(false, Ag[1], false, B1, (short)0, cg, false, false);
    cf = __builtin_amdgcn_wmma_f32_16x16x4_f32(false, Af[2], false, B2, (short)0, cf, false, false);
    cg = __builtin_amdgcn_wmma_f32_16x16x4_f32(false, Ag[2], false, B2, (short)0, cg, false, false);
    cf = __builtin_amdgcn_wmma_f32_16x16x4_f32(false, Af[3], false, B3, (short)0, cf, false, false);
    cg = __builtin_amdgcn_wmma_f32_16x16x4_f32(false, Ag[3], false, B3, (short)0, cg, false, false);

    // nonlinearity + state update (hardware v_tanh_f32)
#pragma unroll
    for (int r = 0; r < 8; ++r) {
      float f = 0.5f * TANHF(0.5f * cf[r]) + 0.5f;  // sigmoid via tanh
      float g = TANHF(cg[r]);
      hD[r] = g + f * (hD[r] - g);                  // f*h + (1-f)*g
    }

    // stream h_t out (fire-and-forget; only consumed by kernel C)
#pragma unroll
    for (int r = 0; r < 8; ++r) hp[ofs + r] = hD[r];
  }
}

// ---------------------------------------------------------------------------
// Kernel C (parallel): recompute FIR phases, rotate h, linear readout.
// Grid: SEQ blocks x 256 threads.
// ---------------------------------------------------------------------------
__global__ void __launch_bounds__(256) out_kernel(
    const float* __restrict__ x,
    const float* __restrict__ firI, const float* __restrict__ firQ,
    const float* __restrict__ WoutI, const float* __restrict__ boutI,
    const float* __restrict__ WoutQ, const float* __restrict__ boutQ,
    const float* __restrict__ hW,
    float* __restrict__ out) {
  __shared__ float sFI[VD * WIN], sFQ[VD * WIN];
  __shared__ float sWI[HID], sWQ[HID];
  __shared__ float sbIQ[2];
  int tid = threadIdx.x;
  if (tid < VD * WIN) { sFI[tid] = firI[tid]; sFQ[tid] = firQ[tid]; }
  if (tid < HID) { sWI[tid] = WoutI[tid]; sWQ[tid] = WoutQ[tid]; }
  if (tid == 0) { sbIQ[0] = boutI[0]; sbIQ[1] = boutQ[0]; }
  __syncthreads();

  int s = blockIdx.x;
  int b = tid;

  float If[VD], Qf[VD];
  fir_cs(x, b, s, sFI, sFQ, If, Qf);

  float ct[VD], st[VD];
#pragma unroll
  for (int v = 0; v < VD; ++v) {
    float m = sqrtf(If[v] * If[v] + Qf[v] * Qf[v]) + 1e-8f;
    ct[v] = If[v] / m;
    st[v] = Qf[v] / m;
  }

  const float* hv = hW + ((size_t)s * B_TOT + b) * HID;
  float oI = sbIQ[0], oQ = sbIQ[1];
#pragma unroll
  for (int j = 0; j < HID; ++j) {
    int v = j % VD;  // theta tiling: [0..5, 0..5, 0..3]
    float h = hv[j];
    oI += h * ct[v] * sWI[j];
    oQ += h * st[v] * sWQ[j];
  }

  float* op = out + ((size_t)b * SEQ + s) * 2;
  op[0] = oI - oQ;
  op[1] = oI + oQ;
}

// ---------------------------------------------------------------------------
extern "C" void kernel_launch(void* const* d_in, const int* in_sizes, int n_in,
                              void* d_out, int out_size, void* d_ws, size_t ws_size,
                              hipStream_t stream) {
  const float* x     = (const float*)d_in[0];
  const float* h0    = (const float*)d_in[1];
  const float* firI  = (const float*)d_in[2];
  const float* firQ  = (const float*)d_in[3];
  const float* Wfi   = (const float*)d_in[4];
  const float* bfi   = (const float*)d_in[5];
  const float* Wfh   = (const float*)d_in[6];
  const float* Wgi   = (const float*)d_in[7];
  const float* bgi   = (const float*)d_in[8];
  const float* Wgh   = (const float*)d_in[9];
  const float* WoutI = (const float*)d_in[10];
  const float* boutI = (const float*)d_in[11];
  const float* WoutQ = (const float*)d_in[12];
  const float* boutQ = (const float*)d_in[13];
  float* out = (float*)d_out;

  // workspace: fi | gi | h_seq, each B*S*HID f32 (67 MB)
  const size_t NF = (size_t)B_TOT * SEQ * HID;
  float* fiW = (float*)d_ws;
  float* giW = fiW + NF;
  float* hW  = giW + NF;

  fir_gate_kernel<<<SEQ, 256, 0, stream>>>(x, firI, firQ, Wfi, bfi, Wgi, bgi, fiW, giW);
  scan_kernel<<<NBLK, 32, 0, stream>>>(h0, Wfh, Wgh, fiW, giW, hW);
  out_kernel<<<SEQ, 256, 0, stream>>>(x, firI, firQ, WoutI, boutI, WoutQ, boutQ, hW, out);
}